// SAGPoolPrompt_74852690035343
// MI455X (gfx1250) — compile-verified
//
#include <hip/hip_runtime.h>
#include <math.h>

typedef __attribute__((ext_vector_type(2))) float v2f;
typedef __attribute__((ext_vector_type(4))) float f4v;
typedef __attribute__((ext_vector_type(8))) float v8f;

#define N_NODES 100000
#define N_EDGES 3200000
#define NB      64
#define IN_CH   256
#define NC      4
#define NTILES  (N_NODES / 16)   // 6250 exactly

// ---------------- init / small kernels ----------------

__global__ void init_kernel(float* deg, float* agg, int* counts) {
    int i = blockIdx.x * blockDim.x + threadIdx.x;
    if (i < N_NODES) deg[i] = 0.0f;
    if (i < 4 * N_NODES) agg[i] = 0.0f;
    if (i < NB) counts[i] = 0;
}

__global__ void pesum_kernel(const float* __restrict__ pe, float* __restrict__ pesum) {
    int f = threadIdx.x;   // 256 threads
    pesum[f] = pe[f] + pe[IN_CH + f] + pe[2 * IN_CH + f] + pe[3 * IN_CH + f];
}

__global__ void deg_kernel(const int* __restrict__ ei, float* deg) {
    int e = blockIdx.x * blockDim.x + threadIdx.x;
    if (e >= N_EDGES) return;
    atomicAdd(&deg[ei[N_EDGES + e]], 1.0f);   // col
}

__global__ void count_kernel(const int* __restrict__ batch, int* counts) {
    int n = blockIdx.x * blockDim.x + threadIdx.x;
    if (n >= N_NODES) return;
    atomicAdd(&counts[batch[n]], 1);
}

__global__ void dinv_kernel(float* deg) {
    int n = blockIdx.x * blockDim.x + threadIdx.x;
    if (n >= N_NODES) return;
    deg[n] = rsqrtf(deg[n] + 1.0f);           // in-place: deg -> dinv
}

__global__ void starts_kernel(const int* __restrict__ counts, int* starts, int* kpg) {
    if (threadIdx.x == 0) {
        int acc = 0;
        for (int g = 0; g < NB; ++g) { starts[g] = acc; acc += counts[g]; }
    }
    int g = threadIdx.x;
    if (g < NB) kpg[g] = (counts[g] + 1) >> 1;  // ceil(0.5 * n)
}

// ---------------- XW = (x + pe_sum) @ W via V_WMMA_F32_16X16X4_F32 ----------------
// One wave per 16-node tile.
//  - sx: 16x256 tile, row stride 260 floats -> A-side ds_load_b64 bank-conflict-free.
//  - sw: W zero-padded to [256][16] so the B fragment loads are UNconditional for all
//    32 lanes (no per-iteration s_and_saveexec around the WMMA loop); cols 4..15 = 0.
//    B addressing (k*16 + lane&15): lanes 0-15 and 16-31 hit disjoint 16-bank groups.

__global__ __launch_bounds__(32) void xw_kernel(const float* __restrict__ x,
                                                const float* __restrict__ W,
                                                const float* __restrict__ pesum,
                                                float* __restrict__ xw) {
    __shared__ float sx[16 * 260];
    __shared__ float sw[IN_CH * 16];
    const int tile = blockIdx.x;
    const int lane = threadIdx.x;
    const float* xt = x + (size_t)tile * 16 * IN_CH;

    // prefetch next tile's x rows (global_prefetch_b8): 32 lanes x 512B = 16KB
    if (tile + 1 < NTILES)
        __builtin_prefetch(xt + 16 * IN_CH + lane * 128, 0, 1);

    // stage x + pe_sum (coalesced float4)
    for (int f = lane; f < 1024; f += 32) {       // 1024 float4s = 16*256 floats
        int row = f >> 6;
        int kb  = (f & 63) << 2;
        f4v v = *(const f4v*)(xt + row * IN_CH + kb);
        f4v p = *(const f4v*)(pesum + kb);
        v.x += p.x; v.y += p.y; v.z += p.z; v.w += p.w;
        *(f4v*)(sx + row * 260 + kb) = v;
    }
    // zero-fill padded W image, then scatter real columns (same-wave DS ops are in order)
    for (int idx = lane; idx < IN_CH * 16; idx += 32)
        sw[idx] = 0.0f;
    for (int idx = lane; idx < IN_CH * NC; idx += 32) {
        int k = idx >> 2, c = idx & 3;
        sw[k * 16 + c] = W[idx];
    }
    __syncthreads();

    const int m    = lane & 15;                   // A row (M) == B col (N)
    const int koff = (lane >> 4) << 1;            // K pair: lanes 0-15 -> K0,1; 16-31 -> K2,3
    v8f acc = {};
    for (int k0 = 0; k0 < IN_CH; k0 += 4) {
        v2f a = *(const v2f*)(sx + m * 260 + k0 + koff);
        v2f bf;
        bf.x = sw[(k0 + koff)     * 16 + m];      // unconditional: padded cols read 0.0
        bf.y = sw[(k0 + koff + 1) * 16 + m];
        acc = __builtin_amdgcn_wmma_f32_16x16x4_f32(false, a, false, bf,
                                                    (short)0, acc, false, false);
    }
    // D layout: VGPR r: lanes 0-15 -> (M=r, N=lane), lanes 16-31 -> (M=8+r, N=lane-16)
    if (m < 4) {
        int mbase = (lane < 16) ? 0 : 8;
        #pragma unroll
        for (int r = 0; r < 8; ++r)
            xw[(size_t)(tile * 16 + mbase + r) * NC + m] = acc[r];
    }
}

// ---------------- edge scatter: agg[col] += dinv[row]*dinv[col]*xw[row] ----------------

__global__ void scatter_kernel(const int* __restrict__ ei, const float* __restrict__ dinv,
                               const float* __restrict__ xw, float* agg) {
    int e = blockIdx.x * blockDim.x + threadIdx.x;
    if (e >= N_EDGES) return;
    int r = ei[e];
    int c = ei[N_EDGES + e];
    float nrm = dinv[r] * dinv[c];
    f4v v = *(const f4v*)(xw + (size_t)r * NC);
    atomicAdd(&agg[(size_t)c * NC + 0], nrm * v.x);
    atomicAdd(&agg[(size_t)c * NC + 1], nrm * v.y);
    atomicAdd(&agg[(size_t)c * NC + 2], nrm * v.z);
    atomicAdd(&agg[(size_t)c * NC + 3], nrm * v.w);
}

// ---------------- score = tanh(agg + dinv^2*xw + b) ----------------

__global__ void score_kernel(const float* __restrict__ agg, const float* __restrict__ xw,
                             const float* __restrict__ dinv, const float* __restrict__ b,
                             float* __restrict__ score) {
    int n = blockIdx.x * blockDim.x + threadIdx.x;
    if (n >= N_NODES) return;
    f4v a = *(const f4v*)(agg + (size_t)n * NC);
    f4v w = *(const f4v*)(xw + (size_t)n * NC);
    float d2 = dinv[n] * dinv[n];
    f4v s;
    s.x = tanhf(a.x + d2 * w.x + b[0]);
    s.y = tanhf(a.y + d2 * w.y + b[1]);
    s.z = tanhf(a.z + d2 * w.z + b[2]);
    s.w = tanhf(a.w + d2 * w.w + b[3]);
    *(f4v*)(score + (size_t)n * NC) = s;
}

// ---------------- per-(graph, cluster) top-k via stable rank counting ----------------
// rank_i = #{j : s_j > s_i or (s_j == s_i and j < i)}  (matches stable lexsort(-s))

#define TOPK_CHUNK 2048

__global__ __launch_bounds__(256) void topk_kernel(const float* __restrict__ score,
                                                   const int* __restrict__ counts,
                                                   const int* __restrict__ starts,
                                                   const int* __restrict__ kpg,
                                                   float* __restrict__ mask) {
    __shared__ float sbuf[TOPK_CHUNK];
    const int g  = blockIdx.x >> 2;
    const int cl = blockIdx.x & 3;
    const int n     = counts[g];
    const int start = starts[g];
    const int k     = kpg[g];

    for (int ibase = 0; ibase < n; ibase += 256) {       // uniform outer loops -> safe barriers
        int i = ibase + (int)threadIdx.x;
        bool act = (i < n);
        float si = act ? score[(size_t)(start + i) * NC + cl] : 0.0f;
        int rank = 0;
        for (int jb = 0; jb < n; jb += TOPK_CHUNK) {
            int cend = n - jb; if (cend > TOPK_CHUNK) cend = TOPK_CHUNK;
            __syncthreads();
            for (int t = threadIdx.x; t < cend; t += 256)
                sbuf[t] = score[(size_t)(start + jb + t) * NC + cl];
            __syncthreads();
            if (act) {
                for (int j = 0; j < cend; ++j) {
                    float sj = sbuf[j];
                    int jj = jb + j;
                    rank += (int)((sj > si) | ((sj == si) & (jj < i)));
                }
            }
        }
        if (act) mask[(size_t)(start + i) * NC + cl] = (rank < k) ? 1.0f : 0.0f;
    }
}

// ---------------- out = x + ((mask*score) @ pools_emb) / idx_count, WMMA K=4 ----------------
// 8 waves/block, one 16-node tile per wave; one 16x16x4 f32 WMMA per 16-channel chunk.
// All lane<16 / lane>=16 splits are value/pointer selects (v_cndmask) -> no EXEC
// manipulation around the WMMAs.

__global__ __launch_bounds__(256) void out_kernel(const float* __restrict__ x,
                                                  const float* __restrict__ pe,
                                                  const float* __restrict__ score,
                                                  const float* __restrict__ mask,
                                                  float* __restrict__ out) {
    const int wave = threadIdx.x >> 5;
    const int lane = threadIdx.x & 31;
    const int tile = blockIdx.x * 8 + wave;
    if (tile >= NTILES) return;                 // wave-uniform; no barriers below

    const int m    = lane & 15;
    const int node = tile * 16 + m;
    const bool hi  = (lane >= 16);
    f4v sc = *(const f4v*)(score + (size_t)node * NC);
    f4v mk = *(const f4v*)(mask  + (size_t)node * NC);
    float inv = 1.0f / (1.0f + mk.x + mk.y + mk.z + mk.w);

    v2f a;                                       // A: 16x4, lanes 0-15 K0,1; lanes 16-31 K2,3
    a.x = hi ? (mk.z * sc.z) : (mk.x * sc.x);
    a.y = hi ? (mk.w * sc.w) : (mk.y * sc.y);

    const int mbase = hi ? 8 : 0;
    float invR[8];
    #pragma unroll
    for (int r = 0; r < 8; ++r)                  // lane M (M<16) holds row M's 1/count
        invR[r] = __shfl(inv, mbase + r, 32);

    const float* peK  = pe + (hi ? 2 * IN_CH : 0);   // B rows K (+K+1) for this half-wave
    const float* xrow = x   + (size_t)tile * 16 * IN_CH;
    float*       orow = out + (size_t)tile * 16 * IN_CH;

    for (int f0 = 0; f0 < IN_CH; f0 += 16) {
        v2f bf;                                  // B: 4x16 rows of pools_emb
        bf.x = peK[f0 + m];
        bf.y = peK[IN_CH + f0 + m];
        v8f cz = {};
        v8f d = __builtin_amdgcn_wmma_f32_16x16x4_f32(false, a, false, bf,
                                                      (short)0, cz, false, false);
        #pragma unroll
        for (int r = 0; r < 8; ++r) {
            int row = mbase + r;
            int idx = row * IN_CH + f0 + m;      // lanes 0-15 & 16-31 each write 64B runs
            orow[idx] = xrow[idx] + d[r] * invR[r];
        }
    }
}

// ---------------- launch ----------------

extern "C" void kernel_launch(void* const* d_in, const int* in_sizes, int n_in,
                              void* d_out, int out_size, void* d_ws, size_t ws_size,
                              hipStream_t stream) {
    const float* x     = (const float*)d_in[0];
    const int*   ei    = (const int*)d_in[1];
    const int*   batch = (const int*)d_in[2];
    const float* pe    = (const float*)d_in[3];
    const float* W     = (const float*)d_in[4];
    const float* bias  = (const float*)d_in[5];
    float* out = (float*)d_out;

    float* wsf    = (float*)d_ws;
    float* deg    = wsf;                         // [N]  (becomes dinv in-place)
    float* xw     = wsf + (size_t)N_NODES;       // [N,4]
    float* agg    = wsf + (size_t)5 * N_NODES;   // [N,4]
    float* score  = wsf + (size_t)9 * N_NODES;   // [N,4]
    float* mask   = wsf + (size_t)13 * N_NODES;  // [N,4]
    float* pesum  = wsf + (size_t)17 * N_NODES;  // [256]
    int*   counts = (int*)(pesum + IN_CH);       // [B]
    int*   starts = counts + NB;                 // [B]
    int*   kpg    = starts + NB;                 // [B]

    init_kernel<<<(4 * N_NODES + 255) / 256, 256, 0, stream>>>(deg, agg, counts);
    pesum_kernel<<<1, IN_CH, 0, stream>>>(pe, pesum);
    deg_kernel<<<(N_EDGES + 255) / 256, 256, 0, stream>>>(ei, deg);
    count_kernel<<<(N_NODES + 255) / 256, 256, 0, stream>>>(batch, counts);
    dinv_kernel<<<(N_NODES + 255) / 256, 256, 0, stream>>>(deg);
    starts_kernel<<<1, 64, 0, stream>>>(counts, starts, kpg);
    xw_kernel<<<NTILES, 32, 0, stream>>>(x, W, pesum, xw);
    scatter_kernel<<<(N_EDGES + 255) / 256, 256, 0, stream>>>(ei, deg, xw, agg);
    score_kernel<<<(N_NODES + 255) / 256, 256, 0, stream>>>(agg, xw, deg, bias, score);
    topk_kernel<<<NB * NC, 256, 0, stream>>>(score, counts, starts, kpg, mask);
    out_kernel<<<(NTILES + 7) / 8, 256, 0, stream>>>(x, pe, score, mask, out);
}